// GroupGenerator_64424509440061
// MI455X (gfx1250) — compile-verified
//
#include <hip/hip_runtime.h>
#include <hip/hip_bf16.h>
#include <math.h>

// ---------------------------------------------------------------------------
// Problem: n = 1024, C = 16 channels (2*8), O = 32 hidden channels.
// u[o,i] = sum_c W1[o,c] * v_abs[c,i]            (32 x 1024)
// h[o,i,j] = relu(u[o,i] - u[o,j] + b1[o])       (never materialized)
// BN stats per o over (i,j); fold BN+W2 into a[o], c.
// draw[i,j] = exp( sum_o a[o]*relu(u[o,i]-u[o,j]+b1[o]) + c )
// dist = 0.5*(draw + draw^T); sig = sigmoid(-(dist-1.0)/0.1)
// indices: sequential row-scan label merge on dist (1 workgroup)
// v_soft = (v @ sig) / colsum  -> WMMA f32 16x16x4 GEMM, epilogue does
// v_out = (v - v_soft) + v_soft (straight-through forward).
// ---------------------------------------------------------------------------

typedef float v2f __attribute__((ext_vector_type(2)));
typedef float v8f __attribute__((ext_vector_type(8)));

#define N 1024
#define TH_C 1.0f
#define INV_TAU 10.0f

// workspace layout (float offsets)
#define U_OFF      0           // 32*1024
#define PART_OFF   32768       // 1024*64  (per-row partial sum/sumsq)
#define COEF_OFF   98304       // a[0..31], c at [32]
#define COLSUM_OFF 98368       // 1024
#define DRAW_OFF   131072      // 1024*1024
#define DIST_OFF   1179648     // 1024*1024
#define SIG_OFF    2228224     // 1024*1024

// ---- u = W1 @ v_abs --------------------------------------------------------
__global__ void k_u(const float* __restrict__ vabs, const float* __restrict__ W1,
                    float* __restrict__ u) {
  int i = blockIdx.x * 256 + threadIdx.x;   // column
  int o = blockIdx.y;                       // output channel
  float acc = 0.f;
  #pragma unroll
  for (int c = 0; c < 16; ++c)
    acc += W1[o * 16 + c] * vabs[c * N + i];
  u[o * N + i] = acc;
}

// ---- per-row partial BN stats (deterministic two-stage reduction) ----------
__global__ void k_stats_partial(const float* __restrict__ u,
                                const float* __restrict__ b1,
                                float* __restrict__ part) {
  int i = blockIdx.x;
  int t = threadIdx.x;          // 256 threads
  int o = t >> 3;               // 0..31
  int s = t & 7;                // 8 j-lanes per channel
  float ui = u[o * N + i];
  float bb = b1[o];
  const float* uo = u + o * N;
  float sum = 0.f, sq = 0.f;
  for (int j = s; j < N; j += 8) {
    float val = fmaxf(ui - uo[j] + bb, 0.f);
    sum += val;
    sq  += val * val;
  }
  // reduce across the 8 consecutive lanes sharing o (stay inside wave32)
  #pragma unroll
  for (int off = 1; off < 8; off <<= 1) {
    sum += __shfl_xor(sum, off, 32);
    sq  += __shfl_xor(sq,  off, 32);
  }
  if (s == 0) {
    part[i * 64 + o]      = sum;
    part[i * 64 + 32 + o] = sq;
  }
}

// ---- finalize stats, fold BN + W2 into a[o], c -----------------------------
__global__ void k_stats_final(const float* __restrict__ part,
                              const float* __restrict__ W2,
                              const float* __restrict__ gamma,
                              const float* __restrict__ beta,
                              const float* __restrict__ b2,
                              float* __restrict__ coef) {
  int o = threadIdx.x;          // 32 threads, one wave
  float S = 0.f, Q = 0.f;
  for (int i = 0; i < N; ++i) {
    S += part[i * 64 + o];
    Q += part[i * 64 + 32 + o];
  }
  const float inv_nn = 1.0f / ((float)N * (float)N);
  float mu  = S * inv_nn;
  float var = Q * inv_nn - mu * mu;
  var = fmaxf(var, 0.f);
  float inv = rsqrtf(var + 1e-5f);
  float a   = W2[o] * gamma[o] * inv;
  float trm = W2[o] * beta[o] - a * mu;
  coef[o] = a;
  #pragma unroll
  for (int off = 16; off > 0; off >>= 1) trm += __shfl_xor(trm, off, 32);
  if (o == 0) coef[32] = trm + b2[0];
}

// ---- fused draw[i,j] = exp(sum_o a*relu(...) + c) --------------------------
__global__ void k_draw(const float* __restrict__ u, const float* __restrict__ b1,
                       const float* __restrict__ coef, float* __restrict__ draw) {
  int i = blockIdx.y;
  int j = blockIdx.x * 256 + threadIdx.x;
  float acc = coef[32];
  #pragma unroll 8
  for (int o = 0; o < 32; ++o) {
    float val = u[o * N + i] - u[o * N + j] + b1[o];
    acc += coef[o] * fmaxf(val, 0.f);
  }
  draw[i * N + j] = expf(acc);
}

// ---- dist = 0.5*(d + d^T); sig = sigmoid(-(dist-th)/tau) -------------------
__global__ void k_sym_sig(const float* __restrict__ draw,
                          float* __restrict__ dist, float* __restrict__ sig) {
  int i = blockIdx.y;
  int j = blockIdx.x * 256 + threadIdx.x;
  float d = 0.5f * (draw[i * N + j] + draw[j * N + i]);
  dist[i * N + j] = d;
  sig[i * N + j]  = 1.0f / (1.0f + expf((d - TH_C) * INV_TAU));
}

// ---- colsum[j] = sum_i sig[i,j] --------------------------------------------
__global__ void k_colsum(const float* __restrict__ sig, float* __restrict__ colsum) {
  int j = blockIdx.x * 256 + threadIdx.x;
  float s = 0.f;
  for (int i = 0; i < N; ++i) s += sig[i * N + j];
  colsum[j] = s;
}

// ---- sequential label-merge scan + rank relabel (1 block, 1024 threads) ----
__global__ void k_labels(const float* __restrict__ dist, float* __restrict__ out_idx) {
  __shared__ int   labels[N];
  __shared__ int   edgef[N];
  __shared__ int   red[32];
  __shared__ int   present[N];
  __shared__ int   rankv[N];
  int t = threadIdx.x;
  labels[t]  = t;
  present[t] = 0;
  __syncthreads();
  for (int r = 1; r < N; ++r) {
    int e = (t < r) && (dist[r * N + t] <= TH_C);
    edgef[t] = e;
    int cand = e ? t : -1;
    #pragma unroll
    for (int off = 16; off > 0; off >>= 1)
      cand = max(cand, __shfl_xor(cand, off, 32));
    if ((t & 31) == 0) red[t >> 5] = cand;
    __syncthreads();
    if (t < 32) {
      int c2 = red[t];
      #pragma unroll
      for (int off = 16; off > 0; off >>= 1)
        c2 = max(c2, __shfl_xor(c2, off, 32));
      if (t == 0) red[0] = c2;
    }
    __syncthreads();
    int cmax = red[0];
    if (cmax >= 0) {
      int L = labels[t];
      // maskS[x] = (x == r) || (E[r,x] && x != cmax)
      if (L == r || (edgef[L] && L != cmax)) labels[t] = cmax;
    }
    __syncthreads();
  }
  present[labels[t]] = 1;
  __syncthreads();
  if (t == 0) {
    int acc = 0;
    for (int k = 0; k < N; ++k) { acc += present[k]; rankv[k] = acc - 1; }
  }
  __syncthreads();
  out_idx[t] = (float)rankv[labels[t]];
}

// ---- WMMA GEMM: v_soft = (v @ sig)/colsum; out = (v - vs) + vs -------------
// One wave per 16-column tile of C (M=16 channels), K=1024 in steps of 4.
// A frag (16x4 f32): lane l<16 -> A[l, 0..1], lane l>=16 -> A[l-16, 2..3].
// B frag (4x16 f32): vgpr0 = rows {k, k+2} across lanes, vgpr1 = rows {k+1, k+3}.
__global__ void k_group_gemm(const float* __restrict__ v,
                             const float* __restrict__ sig,
                             const float* __restrict__ colsum,
                             float* __restrict__ out) {
  int wave = threadIdx.x >> 5;                 // 8 waves per block
  int lane = threadIdx.x & 31;
  int half = lane >> 4;                        // 0 or 1
  int l15  = lane & 15;
  int n0   = (blockIdx.x * 8 + wave) * 16;     // 64 tiles cover N=1024
  const float* arow = v + l15 * N + 2 * half;  // row m = l15, k offset 2*half
  v8f c = {};
  for (int k = 0; k < N; k += 4) {
    v2f a, b;
    a[0] = arow[k];
    a[1] = arow[k + 1];
    b[0] = sig[(k + 2 * half)     * N + n0 + l15];
    b[1] = sig[(k + 2 * half + 1) * N + n0 + l15];
    c = __builtin_amdgcn_wmma_f32_16x16x4_f32(
        /*neg_a=*/false, a, /*neg_b=*/false, b,
        /*c_mod=*/(short)0, c, /*reuse_a=*/false, /*reuse_b=*/false);
  }
  int col = n0 + l15;
  float inv_cs = 1.0f / colsum[col];
  #pragma unroll
  for (int r = 0; r < 8; ++r) {
    int m = r + 8 * half;                      // C layout: vgpr r -> M=r / M=r+8
    float vs = c[r] * inv_cs;
    float vv = v[m * N + col];
    out[m * N + col] = (vv - vs) + vs;         // straight-through forward
  }
}

extern "C" void kernel_launch(void* const* d_in, const int* in_sizes, int n_in,
                              void* d_out, int out_size, void* d_ws, size_t ws_size,
                              hipStream_t stream) {
  const float* v     = (const float*)d_in[0];   // [1,2,8,1024] -> [16][1024]
  const float* vabs  = (const float*)d_in[1];
  const float* W1    = (const float*)d_in[2];   // [32,16]
  const float* b1    = (const float*)d_in[3];
  const float* gamma = (const float*)d_in[4];
  const float* beta  = (const float*)d_in[5];
  const float* W2    = (const float*)d_in[6];   // [1,32]
  const float* b2    = (const float*)d_in[7];
  float* out = (float*)d_out;                   // [16*1024 v_out | 1024 indices]
  float* ws  = (float*)d_ws;

  float* u      = ws + U_OFF;
  float* part   = ws + PART_OFF;
  float* coef   = ws + COEF_OFF;
  float* colsum = ws + COLSUM_OFF;
  float* draw   = ws + DRAW_OFF;
  float* dist   = ws + DIST_OFF;
  float* sig    = ws + SIG_OFF;

  k_u<<<dim3(N / 256, 32), 256, 0, stream>>>(vabs, W1, u);
  k_stats_partial<<<N, 256, 0, stream>>>(u, b1, part);
  k_stats_final<<<1, 32, 0, stream>>>(part, W2, gamma, beta, b2, coef);
  k_draw<<<dim3(N / 256, N), 256, 0, stream>>>(u, b1, coef, draw);
  k_sym_sig<<<dim3(N / 256, N), 256, 0, stream>>>(draw, dist, sig);
  k_colsum<<<N / 256, 256, 0, stream>>>(sig, colsum);
  k_labels<<<1, N, 0, stream>>>(dist, out + 16 * N);
  k_group_gemm<<<8, 256, 0, stream>>>(v, sig, colsum, out);
}